// Bank_42253888258377
// MI455X (gfx1250) — compile-verified
//
#include <hip/hip_runtime.h>

// ---------------------------------------------------------------------------
// Problem constants (match reference)
// ---------------------------------------------------------------------------
#define NUM_CLASSES 1000
#define FEAT        64
#define WARMUP_STEP 1000
#define LAM         0.9f

// Accumulator: 64 sums + 1 count per class, stride 65 so that within a wave
// (32 consecutive dims) LDS banks (addr/4 % 64) are all distinct -> conflict-
// free ds_add_f32, and the count column is spread across banks by class.
#define ACC_STRIDE  65
#define ACC_ELEMS   (NUM_CLASSES * ACC_STRIDE)      // 65,000 floats = 260,000 B

// TDM streaming chunk: 96 rows x 64 f32 = 24 KiB, double buffered.
#define CHUNK_ROWS  96
#define CHUNK_ELEMS (CHUNK_ROWS * FEAT)             // 6,144 floats
#define P1_BLOCKS   128
#define P1_THREADS  1024                            // 32 waves (wave32)
#define ELEMS_PER_THREAD (CHUNK_ELEMS / P1_THREADS) // 6
#define MAX_PART    128

// LDS budget: 260,000 (acc) + 2*24,576 (stage) + 384 (labels) = 309,536 B
// <= 320 KiB per workgroup (CDNA5 WGP LDS). One workgroup per WGP.
#define P1_SMEM_BYTES ((ACC_ELEMS + 2 * CHUNK_ELEMS) * 4 + CHUNK_ROWS * 4)

typedef unsigned int v4u __attribute__((ext_vector_type(4)));
typedef int          v8i __attribute__((ext_vector_type(8)));
typedef int          v4i __attribute__((ext_vector_type(4)));

// ---------------------------------------------------------------------------
// Tensor Data Mover: 1D-tile load of CHUNK_ELEMS f32 from global -> LDS.
// D# packing per CDNA5 ISA ch.8:
//   group0: [1:0]=count=1, [63:32]=lds_addr, [120:64]=global_addr, [127:126]=2
//   group1: data_size=2 (4B) @[17:16]; tensor_dim0 @[79:48] (OOB reads -> 0);
//           tensor_dim1=1 @[111:80]; tile_dim0=CHUNK_ELEMS @[127:112];
//           tile_dim1/2 = 0 (unused); tensor_dim0_stride @[207:160].
//   groups 2/3: zero (2D descriptor, iterate/gather disabled).
// This toolchain (clang-23 / therock-10.0 headers) uses the 6-arg builtin:
//   (uint32x4 g0, int32x8 g1, int32x4, int32x4, int32x8, i32 cpol)
// Completion tracked with TENSORcnt (s_wait_tensorcnt).
// ---------------------------------------------------------------------------
__device__ __forceinline__ void tdm_load_chunk(unsigned lds_byte_addr,
                                               const float* gsrc,
                                               unsigned long long remaining_elems) {
  unsigned long long ga = (unsigned long long)(const void*)gsrc;
  unsigned td0 = remaining_elems > 0xFFFFFFFFull ? 0xFFFFFFFFu
                                                 : (unsigned)remaining_elems;
  v4u g0;
  g0[0] = 1u;                                              // count=1, user D#
  g0[1] = lds_byte_addr;                                   // LDS dest (bytes)
  g0[2] = (unsigned)ga;                                    // global addr lo
  g0[3] = ((unsigned)(ga >> 32) & 0x01FFFFFFu) | (2u << 30); // addr hi | type=2
  v8i g1;
  g1[0] = (int)(2u << 16);                                 // data_size = 4 B
  g1[1] = (int)((td0 & 0xFFFFu) << 16);                    // tensor_dim0 lo16
  g1[2] = (int)((td0 >> 16) | (1u << 16));                 // dim0 hi16 | dim1=1
  g1[3] = (int)((unsigned)CHUNK_ELEMS << 16);              // tile_dim0
  g1[4] = 0;                                               // tile_dim1/2 unused
  g1[5] = (int)td0;                                        // dim0_stride lo32
  g1[6] = 0;                                               // stride hi | d1str
  g1[7] = 0;
  v4i gz  = (v4i)0;
  v8i gz8 = (v8i)0;
  __builtin_amdgcn_tensor_load_to_lds(g0, g1, gz, gz, gz8, 0);
}

// ---------------------------------------------------------------------------
// Phase 0: zero the workspace partial accumulators (graph-capture safe).
// ---------------------------------------------------------------------------
__global__ void proto_zero_ws(float* __restrict__ ws, long long n) {
  long long i = (long long)blockIdx.x * blockDim.x + threadIdx.x;
  long long stride = (long long)gridDim.x * blockDim.x;
  for (; i < n; i += stride) ws[i] = 0.0f;
}

// ---------------------------------------------------------------------------
// Phase 1: TDM-streamed, LDS-privatized scatter-add of features into
// per-workgroup [1000][65] accumulators; flush with global f32 atomics.
// ---------------------------------------------------------------------------
__global__ void __launch_bounds__(P1_THREADS)
proto_scatter_accum(const float* __restrict__ feature,
                    const int*   __restrict__ label,
                    float*       __restrict__ ws,
                    long long nrows, int npart) {
  extern __shared__ char smem_raw[];
  float* acc    = (float*)smem_raw;             // [1000][65]
  float* stage0 = acc + ACC_ELEMS;              // [6144]
  float* stage1 = stage0 + CHUNK_ELEMS;         // [6144]
  int*   lab    = (int*)(stage1 + CHUNK_ELEMS); // [96]

  const int tid = threadIdx.x;
  long long rpb  = (nrows + P1_BLOCKS - 1) / P1_BLOCKS;
  long long rBeg = (long long)blockIdx.x * rpb;
  long long rEnd = rBeg + rpb; if (rEnd > nrows) rEnd = nrows;
  long long rowsBlk = (rEnd > rBeg) ? (rEnd - rBeg) : 0;
  const int nchunks = (int)((rowsBlk + CHUNK_ROWS - 1) / CHUNK_ROWS);
  const long long totElems = nrows * FEAT;

  // Issue chunk 0 DMA immediately; it overlaps the accumulator zero-init.
  if (tid == 0 && nchunks > 0) {
    tdm_load_chunk((unsigned)(unsigned long long)(void*)stage0,
                   feature + rBeg * FEAT,
                   (unsigned long long)(totElems - rBeg * FEAT));
  }

  // Software-pipelined label prefetch (one chunk ahead in registers).
  int curLab = 0, nextLab = 0;
  if (tid < CHUNK_ROWS && nchunks > 0) {
    long long r = rBeg + tid;
    curLab = (r < rEnd) ? label[r] : 0;
  }

  for (int i = tid; i < ACC_ELEMS; i += P1_THREADS) acc[i] = 0.0f;

  for (int c = 0; c < nchunks; ++c) {
    const long long cb = rBeg + (long long)c * CHUNK_ROWS;
    const long long nb = cb + CHUNK_ROWS;
    float* stage  = (c & 1) ? stage1 : stage0;
    float* nstage = (c & 1) ? stage0 : stage1;

    if (tid == 0) {
      if (c + 1 < nchunks) {
        tdm_load_chunk((unsigned)(unsigned long long)(void*)nstage,
                       feature + nb * FEAT,
                       (unsigned long long)(totElems - nb * FEAT));
        __builtin_amdgcn_s_wait_tensorcnt(1);  // chunk c done, c+1 in flight
      } else {
        __builtin_amdgcn_s_wait_tensorcnt(0);  // drain last chunk
      }
    }
    if (tid < CHUNK_ROWS) {
      lab[tid] = curLab;                       // publish labels for chunk c
      if (c + 1 < nchunks) {                   // prefetch labels for chunk c+1
        long long r = nb + tid;
        nextLab = (r < rEnd) ? label[r] : 0;
      }
    }
    __syncthreads();

    long long vrows = rEnd - cb; if (vrows > CHUNK_ROWS) vrows = CHUNK_ROWS;
    const int validElems = (int)vrows * FEAT;
    #pragma unroll
    for (int k = 0; k < ELEMS_PER_THREAD; ++k) {
      int e = tid + k * P1_THREADS;
      if (e < validElems) {
        int   d   = e & (FEAT - 1);
        int   cls = lab[e >> 6];
        float v   = stage[e];
        __hip_atomic_fetch_add(&acc[cls * ACC_STRIDE + d], v,
                               __ATOMIC_RELAXED, __HIP_MEMORY_SCOPE_WORKGROUP);
        if (d == 0)  // one count bump per row
          __hip_atomic_fetch_add(&acc[cls * ACC_STRIDE + FEAT], 1.0f,
                                 __ATOMIC_RELAXED, __HIP_MEMORY_SCOPE_WORKGROUP);
      }
    }
    __syncthreads();   // protect stage/lab buffers before reuse
    curLab = nextLab;
  }

  // Flush this workgroup's partial into one of npart global accumulators.
  float* dst = ws + (size_t)(blockIdx.x % npart) * ACC_ELEMS;
  for (int i = tid; i < ACC_ELEMS; i += P1_THREADS) {
    float v = acc[i];
    if (v != 0.0f)
      __hip_atomic_fetch_add(&dst[i], v,
                             __ATOMIC_RELAXED, __HIP_MEMORY_SCOPE_AGENT);
  }
}

// ---------------------------------------------------------------------------
// Phase 2: reduce partials, compute means, apply EMA / warmup / present logic.
// ---------------------------------------------------------------------------
__global__ void proto_finalize(const float* __restrict__ proto,
                               const float* __restrict__ ws,
                               const int*   __restrict__ stepPtr,
                               float*       __restrict__ out, int npart) {
  const int c = blockIdx.x;   // class
  const int d = threadIdx.x;  // 0..63
  __shared__ float s_cnt;
  __shared__ int   s_nz;
  if (d == 0) {
    float cnt = 0.0f;
    for (int p = 0; p < npart; ++p)
      cnt += ws[(size_t)p * ACC_ELEMS + c * ACC_STRIDE + FEAT];
    s_cnt = cnt;
    s_nz  = 0;
  }
  __syncthreads();
  float s = 0.0f;
  for (int p = 0; p < npart; ++p)
    s += ws[(size_t)p * ACC_ELEMS + c * ACC_STRIDE + d];
  float pv = proto[c * FEAT + d];
  if (pv != 0.0f) atomicOr(&s_nz, 1);
  __syncthreads();

  const float cnt     = s_cnt;
  const bool  present = cnt > 0.0f;
  const float mean    = s / fmaxf(cnt, 1.0f);
  const bool  use_new = (s_nz == 0) || (*stepPtr <= WARMUP_STEP);
  const float upd     = use_new ? mean : (LAM * pv + (1.0f - LAM) * mean);
  out[c * FEAT + d]   = present ? upd : pv;
}

// ---------------------------------------------------------------------------
// Host launcher
// ---------------------------------------------------------------------------
extern "C" void kernel_launch(void* const* d_in, const int* in_sizes, int n_in,
                              void* d_out, int out_size, void* d_ws, size_t ws_size,
                              hipStream_t stream) {
  const float* proto   = (const float*)d_in[0];   // [1000,64]
  const float* feature = (const float*)d_in[1];   // [N,64]
  const int*   label   = (const int*)d_in[2];     // [N]
  const int*   step    = (const int*)d_in[3];     // scalar
  float*       out     = (float*)d_out;           // [1000,64]
  float*       ws      = (float*)d_ws;

  long long N = (long long)in_sizes[1] / FEAT;

  // As many independent partial accumulators as the workspace allows
  // (fewer partials -> more global-atomic contention, still correct).
  size_t per = (size_t)ACC_ELEMS * sizeof(float);
  int npart = (int)(ws_size / per);
  if (npart < 1) npart = 1;
  if (npart > MAX_PART) npart = MAX_PART;

  long long zeroN = (long long)npart * ACC_ELEMS;
  int zb = (int)((zeroN + 1023) / 1024);
  if (zb > 1024) zb = 1024;
  proto_zero_ws<<<zb, 1024, 0, stream>>>(ws, zeroN);

  proto_scatter_accum<<<P1_BLOCKS, P1_THREADS, P1_SMEM_BYTES, stream>>>(
      feature, label, ws, N, npart);

  proto_finalize<<<NUM_CLASSES, FEAT, 0, stream>>>(proto, ws, step, out, npart);
}